// GCN_2_layers_64991445123883
// MI455X (gfx1250) — compile-verified
//
#include <hip/hip_runtime.h>

typedef __attribute__((ext_vector_type(2))) float v2f;
typedef __attribute__((ext_vector_type(8))) float v8f;

// ---------------- degree / normalization ----------------
__global__ void deg_init_kernel(float* deg, int n) {
  int i = blockIdx.x * blockDim.x + threadIdx.x;
  if (i < n) deg[i] = 1.0f;  // self-loop contributes 1
}

__global__ void deg_accum_kernel(const int* __restrict__ dst, float* deg, int E) {
  int e = blockIdx.x * blockDim.x + threadIdx.x;
  if (e < E) atomicAdd(&deg[dst[e]], 1.0f);
}

__global__ void deg_rsqrt_kernel(float* deg, int n) {
  int i = blockIdx.x * blockDim.x + threadIdx.x;
  if (i < n) deg[i] = rsqrtf(deg[i]);  // deg >= 1 always (self-loops)
}

// ---------------- WMMA fp32 GEMM: C[n,NOUT] = A[n,K] @ B[K,NOUT] ----------------
// One wave32 computes one 16x16 tile of C using V_WMMA_F32_16X16X4_F32.
// A frag (16x4): lane m (0-15): v0=A[m][k], v1=A[m][k+1]; lane 16+m: v0=A[m][k+2], v1=A[m][k+3]
// B frag (4x16): v0: lanes0-15 row k (N=lane), lanes16-31 row k+2; v1: rows k+1 / k+3
// D (16x16):    VGPR r: lanes0-15 = D[r][lane], lanes16-31 = D[r+8][lane-16]
template <int K, int NOUT>
__global__ void gemm_wmma_kernel(const float* __restrict__ A,
                                 const float* __restrict__ B,
                                 float* __restrict__ C, int n) {
  constexpr int NTILES = NOUT / 16;
  const int wave = blockIdx.x * (blockDim.x >> 5) + (threadIdx.x >> 5);
  const int lane = threadIdx.x & 31;
  const int mtile = wave / NTILES;
  const int ntile = wave % NTILES;
  const int m0 = mtile * 16;
  if (m0 >= n) return;  // uniform across wave

  const int l15 = lane & 15;
  const int khalf = lane >> 4;  // 0: K even pair base, 1: +2
  int arow = m0 + l15;
  if (arow >= n) arow = n - 1;  // clamp loads; store is guarded (keeps EXEC all-1 for WMMA)
  const float* __restrict__ Ap = A + (size_t)arow * K;
  const float* __restrict__ Bp = B + ntile * 16 + l15;

  v8f acc = {};
#pragma unroll
  for (int k = 0; k < K / 4; ++k) {
    const int kk = k * 4 + khalf * 2;
    v2f a, b;
    a.x = Ap[kk];
    a.y = Ap[kk + 1];
    b.x = Bp[(size_t)kk * NOUT];
    b.y = Bp[(size_t)(kk + 1) * NOUT];
    acc = __builtin_amdgcn_wmma_f32_16x16x4_f32(
        /*neg_a=*/false, a, /*neg_b=*/false, b,
        /*c_mod=*/(short)0, acc, /*reuse_a=*/false, /*reuse_b=*/false);
  }

#pragma unroll
  for (int r = 0; r < 8; ++r) {
    const int row = m0 + r + khalf * 8;
    if (row < n) C[(size_t)row * NOUT + ntile * 16 + l15] = acc[r];
  }
}

// ---------------- self-loop + bias init: out[i,f] = bias[f] + feat[i,f]*dis[i]^2 ----
template <int LOGF>
__global__ void self_bias_init_kernel(const float* __restrict__ feat,
                                      const float* __restrict__ dis,
                                      const float* __restrict__ bias,
                                      float* __restrict__ out, int n) {
  constexpr int F = 1 << LOGF;
  const size_t t = (size_t)blockIdx.x * blockDim.x + threadIdx.x;
  if (t >= (size_t)n << LOGF) return;
  const int i = (int)(t >> LOGF);
  const int f = (int)(t & (F - 1));
  const float d = dis[i];
  out[t] = bias[f] + feat[t] * d * d;
}

// ---------------- edge scatter: out[dst,f] += feat[src,f]*dis[src]*dis[dst] --------
template <int LOGF>
__global__ void edge_scatter_kernel(const int* __restrict__ src,
                                    const int* __restrict__ dst,
                                    const float* __restrict__ dis,
                                    const float* __restrict__ feat,
                                    float* out, int E) {
  constexpr int F = 1 << LOGF;
  const size_t t = (size_t)blockIdx.x * blockDim.x + threadIdx.x;
  if (t >= (size_t)E << LOGF) return;
  const int e = (int)(t >> LOGF);
  const int f = (int)(t & (F - 1));
  const int s = src[e];
  const int d = dst[e];
  const float w = dis[s] * dis[d];
  atomicAdd(&out[(size_t)d << LOGF | f], feat[(size_t)s << LOGF | f] * w);
}

__global__ void relu_kernel(float* h, size_t total) {
  const size_t t = (size_t)blockIdx.x * blockDim.x + threadIdx.x;
  if (t < total) h[t] = fmaxf(h[t], 0.0f);
}

// ---------------- launch ----------------
extern "C" void kernel_launch(void* const* d_in, const int* in_sizes, int n_in,
                              void* d_out, int out_size, void* d_ws, size_t ws_size,
                              hipStream_t stream) {
  const float* x  = (const float*)d_in[0];
  const int*   ei = (const int*)d_in[1];
  const float* W1 = (const float*)d_in[2];
  const float* b1 = (const float*)d_in[3];
  const float* W2 = (const float*)d_in[4];
  const float* b2 = (const float*)d_in[5];
  float* out = (float*)d_out;

  const int n = in_sizes[0] / 128;  // 100000
  const int E = in_sizes[1] / 2;    // 1600000
  const int* src = ei;
  const int* dst = ei + E;

  // workspace layout (fp32): dis[n] | xw[n*64] | h[n*64] | hw[n*16]
  float* dis = (float*)d_ws;
  float* xw  = dis + n;
  float* h   = xw + (size_t)n * 64;
  float* hw  = h  + (size_t)n * 64;

  const int T = 256;
  auto cdiv = [](size_t a, size_t b) { return (int)((a + b - 1) / b); };

  // degrees -> dis
  deg_init_kernel<<<cdiv(n, T), T, 0, stream>>>(dis, n);
  deg_accum_kernel<<<cdiv(E, T), T, 0, stream>>>(dst, dis, E);
  deg_rsqrt_kernel<<<cdiv(n, T), T, 0, stream>>>(dis, n);

  // layer 1: xw = x @ W1  (K=128, NOUT=64) -> 16x16 tiles, 4 ntiles
  {
    const int mtiles = (n + 15) / 16;
    const int waves = mtiles * 4;
    gemm_wmma_kernel<128, 64><<<cdiv((size_t)waves * 32, T), T, 0, stream>>>(x, W1, xw, n);
  }
  self_bias_init_kernel<6><<<cdiv((size_t)n << 6, T), T, 0, stream>>>(xw, dis, b1, h, n);
  edge_scatter_kernel<6><<<cdiv((size_t)E << 6, T), T, 0, stream>>>(src, dst, dis, xw, h, E);
  relu_kernel<<<cdiv((size_t)n << 6, T), T, 0, stream>>>(h, (size_t)n << 6);

  // layer 2: hw = h @ W2  (K=64, NOUT=16) -> 1 ntile
  {
    const int mtiles = (n + 15) / 16;
    gemm_wmma_kernel<64, 16><<<cdiv((size_t)mtiles * 32, T), T, 0, stream>>>(h, W2, hw, n);
  }
  self_bias_init_kernel<4><<<cdiv((size_t)n << 4, T), T, 0, stream>>>(hw, dis, b2, out, n);
  edge_scatter_kernel<4><<<cdiv((size_t)E << 4, T), T, 0, stream>>>(src, dst, dis, hw, out, E);
}